// MGNNI_m_att_65412351918550
// MI455X (gfx1250) — compile-verified
//
#include <hip/hip_runtime.h>
#include <math.h>

typedef float v2f __attribute__((ext_vector_type(2)));
typedef float v8f __attribute__((ext_vector_type(8)));

#define MF 128   // feature dim
#define MY 10    // output classes

// ---------------- degree count ----------------
__global__ void k_degree(const int* __restrict__ src, const int* __restrict__ dst,
                         float* __restrict__ degs, float* __restrict__ degd, int E) {
    int e = blockIdx.x * blockDim.x + threadIdx.x;
    if (e < E) {
        atomicAdd(&degs[src[e]], 1.0f);
        atomicAdd(&degd[dst[e]], 1.0f);
    }
}

// ---------------- symmetric-normalized edge weights ----------------
__global__ void k_normw(const int* __restrict__ src, const int* __restrict__ dst,
                        const float* __restrict__ ew,
                        const float* __restrict__ degs, const float* __restrict__ degd,
                        float* __restrict__ w, int E) {
    int e = blockIdx.x * blockDim.x + threadIdx.x;
    if (e < E) {
        float ds = degs[src[e]], dd = degd[dst[e]];
        float is = ds > 0.0f ? rsqrtf(ds) : 0.0f;
        float id = dd > 0.0f ? rsqrtf(dd) : 0.0f;
        w[e] = is * ew[e] * id;
    }
}

// ---------------- X [M,N] -> Xn [N,M] (tiled transpose) ----------------
__global__ void k_transpose(const float* __restrict__ X, float* __restrict__ Xn, int N) {
    __shared__ float tile[32][33];
    int i0 = blockIdx.x * 32;     // node tile
    int m0 = blockIdx.y * 32;     // feature tile (128 divisible by 32)
    int tx = threadIdx.x, ty = threadIdx.y;   // (32, 8)
    for (int r = 0; r < 32; r += 8) {
        int m = m0 + ty + r;
        int i = i0 + tx;
        tile[ty + r][tx] = (i < N) ? X[(size_t)m * N + i] : 0.0f;
    }
    __syncthreads();
    for (int r = 0; r < 32; r += 8) {
        int i = i0 + ty + r;
        int m = m0 + tx;
        if (i < N) Xn[(size_t)i * MF + m] = tile[tx][ty + r];
    }
}

// ---------------- gF = (F^T F) / (||F^T F||_F + eps) ---- single block ----
__global__ void k_gf(const float* __restrict__ F, float* __restrict__ gF) {
    __shared__ float red[256];
    __shared__ float invn;
    int t = threadIdx.x;
    float ss = 0.0f;
    for (int idx = t; idx < MF * MF; idx += 256) {
        int a = idx >> 7, b = idx & (MF - 1);
        float acc = 0.0f;
        for (int m = 0; m < MF; ++m)
            acc = fmaf(F[m * MF + a], F[m * MF + b], acc);
        gF[idx] = acc;
        ss += acc * acc;
    }
    red[t] = ss;
    __syncthreads();
    for (int s = 128; s > 0; s >>= 1) {
        if (t < s) red[t] += red[t + s];
        __syncthreads();
    }
    if (t == 0) invn = 1.0f / (sqrtf(red[0]) + 1e-12f);
    __syncthreads();
    float iv = invn;
    for (int idx = t; idx < MF * MF; idx += 256) gF[idx] *= iv;
}

// ---------------- propagation: out[dst] += w * in[src] (out pre-zeroed) ----
__global__ void k_prop(const int* __restrict__ src, const int* __restrict__ dst,
                       const float* __restrict__ w,
                       const float* __restrict__ in, float* __restrict__ out, int E) {
    int lane = threadIdx.x & 31;
    int e = blockIdx.x * (blockDim.x >> 5) + (threadIdx.x >> 5);
    if (e >= E) return;
    int s = src[e], d = dst[e];
    float we = w[e];
    float4 v = ((const float4*)(in + (size_t)s * MF))[lane];
    float* op = out + (size_t)d * MF + lane * 4;
    atomicAdd(op + 0, we * v.x);
    atomicAdd(op + 1, we * v.y);
    atomicAdd(op + 2, we * v.z);
    atomicAdd(op + 3, we * v.w);
}

// ---------------- Z = gamma * (P @ gF^T) + Xn  via V_WMMA_F32_16X16X4_F32 ----
// Node-major: Z[i][j] = gamma * sum_k P[i][k]*gF[j][k] + Xn[i][j]
// Block = 256 threads = 8 waves; wave -> j-tile (8 * 16 = 128), blockIdx.x -> node tile of 16.
__global__ __launch_bounds__(256)
void k_gemm_axpy(const float* __restrict__ P, const float* __restrict__ gF,
                 const float* __restrict__ Xn, const float* __restrict__ gamma,
                 float* __restrict__ Z) {
    int lane = threadIdx.x & 31;
    int wave = threadIdx.x >> 5;
    int i0 = blockIdx.x * 16;
    int j0 = wave * 16;
    int half = lane >> 4;         // 0: K+0/1, 1: K+2/3  (ISA 16x4 f32 A layout)
    int l = lane & 15;
    const float* Arow = P  + (size_t)(i0 + l) * MF + 2 * half;  // A[i][k..]
    const float* Brow = gF + (size_t)(j0 + l) * MF + 2 * half;  // B[k][j] = gF[j][k]
    v8f c = {};
    for (int k = 0; k < MF; k += 4) {
        v2f a = *(const v2f*)(Arow + k);
        v2f b = *(const v2f*)(Brow + k);
        c = __builtin_amdgcn_wmma_f32_16x16x4_f32(false, a, false, b,
                                                  (short)0, c, false, false);
    }
    float g = gamma[0];
    // D layout: lane -> column j = j0 + (lane&15); VGPR v -> row i = i0 + v + 8*half
    for (int v = 0; v < 8; ++v) {
        size_t off = (size_t)(i0 + v + 8 * half) * MF + (j0 + l);
        Z[off] = fmaf(g, c[v], Xn[off]);
    }
}

// scalar tail for node tiles that don't fill 16 (N % 16 != 0)
__global__ void k_gemm_tail(const float* __restrict__ P, const float* __restrict__ gF,
                            const float* __restrict__ Xn, const float* __restrict__ gamma,
                            float* __restrict__ Z, int istart, int N) {
    int i = istart + blockIdx.x;
    int j = threadIdx.x;   // 128 threads
    if (i >= N) return;
    float acc = 0.0f;
    for (int k = 0; k < MF; ++k)
        acc = fmaf(P[(size_t)i * MF + k], gF[(size_t)j * MF + k], acc);
    size_t off = (size_t)i * MF + j;
    Z[off] = fmaf(gamma[0], acc, Xn[off]);
}

// ---------------- attention fusion + final projection (wave per node) ----
__global__ void k_fuse(const float* __restrict__ Z1, const float* __restrict__ Z2,
                       const float* __restrict__ W1, const float* __restrict__ b1,
                       const float* __restrict__ W2, const float* __restrict__ b2,
                       const float* __restrict__ Bm, float* __restrict__ out, int N) {
    int lane = threadIdx.x & 31;
    int node = blockIdx.x * (blockDim.x >> 5) + (threadIdx.x >> 5);
    if (node >= N) return;
    const float* z1 = Z1 + (size_t)node * MF;
    const float* z2 = Z2 + (size_t)node * MF;
    int s = lane >> 4, t = lane & 15;     // lanes 0-15: scale1, 16-31: scale2
    const float* zs = s ? z2 : z1;
    float acc = 0.0f;
    for (int m = 0; m < MF; ++m)
        acc = fmaf(W1[t * MF + m], zs[m], acc);
    float h = tanhf(acc + b1[t]);
    float part = W2[t] * h;
    // reduce within each 16-lane half (xor masks stay inside the half)
    for (int o = 8; o > 0; o >>= 1) part += __shfl_xor(part, o, 32);
    float logit = part + b2[0];
    float lg1 = __shfl(logit, 0, 32);
    float lg2 = __shfl(logit, 16, 32);
    float mx = fmaxf(lg1, lg2);
    float e1 = __expf(lg1 - mx), e2 = __expf(lg2 - mx);
    float inv = 1.0f / (e1 + e2);
    float beta1 = e1 * inv, beta2 = e2 * inv;
    float4 a  = ((const float4*)z1)[lane];
    float4 bb = ((const float4*)z2)[lane];
    float4 f;
    f.x = beta1 * a.x + beta2 * bb.x;
    f.y = beta1 * a.y + beta2 * bb.y;
    f.z = beta1 * a.z + beta2 * bb.z;
    f.w = beta1 * a.w + beta2 * bb.w;
    for (int c = 0; c < MY; ++c) {
        float4 br = ((const float4*)(Bm + c * MF))[lane];
        float p = br.x * f.x + br.y * f.y + br.z * f.z + br.w * f.w;
        for (int o = 16; o > 0; o >>= 1) p += __shfl_xor(p, o, 32);
        if (lane == 0) out[(size_t)node * MY + c] = p;
    }
}

extern "C" void kernel_launch(void* const* d_in, const int* in_sizes, int n_in,
                              void* d_out, int out_size, void* d_ws, size_t ws_size,
                              hipStream_t stream) {
    const float* X  = (const float*)d_in[0];
    const int*   ei = (const int*)d_in[1];
    const float* ew = (const float*)d_in[2];
    const float* F1 = (const float*)d_in[4];
    const float* F2 = (const float*)d_in[5];
    const float* g1 = (const float*)d_in[6];
    const float* g2 = (const float*)d_in[7];
    const float* W1 = (const float*)d_in[8];
    const float* b1 = (const float*)d_in[9];
    const float* W2 = (const float*)d_in[10];
    const float* b2 = (const float*)d_in[11];
    const float* Bm = (const float*)d_in[12];

    const int N = in_sizes[0] / MF;
    const int E = in_sizes[2];
    const int* src = ei;
    const int* dst = ei + E;

    // -------- workspace layout (floats) --------
    float* wsf = (float*)d_ws;
    size_t NM = (size_t)N * MF;
    float* w    = wsf;  wsf += E;
    float* degs = wsf;  wsf += N;
    float* degd = wsf;  wsf += N;
    float* gF   = wsf;  wsf += MF * MF;
    float* Xn   = wsf;  wsf += NM;
    float* P0   = wsf;  wsf += NM;
    float* P1   = wsf;  wsf += NM;
    float* Z1   = wsf;  wsf += NM;
    float* Z2   = wsf;  wsf += NM;

    // -------- graph normalization --------
    hipMemsetAsync(degs, 0, sizeof(float) * N, stream);
    hipMemsetAsync(degd, 0, sizeof(float) * N, stream);
    k_degree<<<(E + 255) / 256, 256, 0, stream>>>(src, dst, degs, degd, E);
    k_normw<<<(E + 255) / 256, 256, 0, stream>>>(src, dst, ew, degs, degd, w, E);

    // -------- X -> node-major --------
    dim3 tg((N + 31) / 32, MF / 32);
    k_transpose<<<tg, dim3(32, 8), 0, stream>>>(X, Xn, N);

    const int ntile = N / 16;
    const int nrem  = N - ntile * 16;

    struct Scale { const float* F; const float* g; int k; float* Z; };
    Scale scales[2] = { {F1, g1, 1, Z1}, {F2, g2, 2, Z2} };

    for (int s = 0; s < 2; ++s) {
        k_gf<<<1, 256, 0, stream>>>(scales[s].F, gF);
        // f^1(0) = X
        hipMemcpyAsync(scales[s].Z, Xn, NM * sizeof(float),
                       hipMemcpyDeviceToDevice, stream);
        // 25 while-loop steps + 1 final pass with identical parameter values
        for (int it = 0; it < 26; ++it) {
            hipMemsetAsync(P0, 0, NM * sizeof(float), stream);
            k_prop<<<(E + 7) / 8, 256, 0, stream>>>(src, dst, w, scales[s].Z, P0, E);
            const float* Pf = P0;
            if (scales[s].k == 2) {
                hipMemsetAsync(P1, 0, NM * sizeof(float), stream);
                k_prop<<<(E + 7) / 8, 256, 0, stream>>>(src, dst, w, P0, P1, E);
                Pf = P1;
            }
            if (ntile > 0)
                k_gemm_axpy<<<ntile, 256, 0, stream>>>(Pf, gF, Xn, scales[s].g,
                                                       scales[s].Z);
            if (nrem > 0)
                k_gemm_tail<<<nrem, MF, 0, stream>>>(Pf, gF, Xn, scales[s].g,
                                                     scales[s].Z, ntile * 16, N);
        }
    }

    // -------- attention fusion + final projection --------
    k_fuse<<<(N + 7) / 8, 256, 0, stream>>>(Z1, Z2, W1, b1, W2, b2, Bm,
                                            (float*)d_out, N);
}